// PCENLeafAdaptation_21844203667492
// MI455X (gfx1250) — compile-verified
//
#include <hip/hip_runtime.h>
#include <cstdint>

typedef __attribute__((ext_vector_type(2))) float v2f;
typedef __attribute__((ext_vector_type(4))) float v4f;
typedef __attribute__((ext_vector_type(8))) float v8f;

#define EPSF   1e-6f
#define T_LEN  16000
#define TILE   256
#define NPASS  63        // 62 full tiles + one 128-element tail

// fast base-2 exp/log: raw hardware ops when clang exposes them
__device__ __forceinline__ float fexp2(float x) {
#if __has_builtin(__builtin_amdgcn_exp2f)
  return __builtin_amdgcn_exp2f(x);      // v_exp_f32
#else
  return exp2f(x);
#endif
}
__device__ __forceinline__ float flog2(float x) {
#if __has_builtin(__builtin_amdgcn_logf)
  return __builtin_amdgcn_logf(x);       // v_log_f32 (base-2)
#else
  return __log2f(x);
#endif
}

extern __shared__ float smem[];   // 8 waves * 512 floats (double-buffered x tiles)

__global__ __launch_bounds__(256) void pcen_wmma_scan_kernel(
    const float* __restrict__ x,
    const float* __restrict__ alpha,
    const float* __restrict__ delta,
    const float* __restrict__ rpow,
    const float* __restrict__ scoef,
    float* __restrict__ out,
    int C, int nrows)
{
  const int lane = threadIdx.x & 31;
  const int wave = threadIdx.x >> 5;
  const int row  = blockIdx.x * 8 + wave;
  if (row >= nrows) return;                      // wave-uniform
  const int ch = row % C;

  // per-channel constants
  const float a_ = alpha[ch];
  const float d_ = delta[ch];
  const float r_ = rpow[ch];
  float s_ = scoef[ch];
  s_ = fminf(fmaxf(s_, 1e-5f), 1.0f);
  const float q     = fmaxf(1.0f - s_, 1e-30f);
  const float inv_s = 1.0f / s_;
  const float L2Q   = flog2(q);
  const float q4    = (q*q)*(q*q);
  const float drc   = fexp2(r_ * flog2(d_));     // delta^r (constant term)

  const float* xrow = x   + (size_t)row * T_LEN;
  float*       orow = out + (size_t)row * T_LEN;

  float* xb0 = smem + wave * 512;                // this wave's double buffer

  const int n    = lane & 15;        // column / row-in-16 index
  const int hi   = lane >> 4;        // lane half
  const int c0   = hi * 2;           // K sub-offset in A/B VGPR striping
  const int eout = n * 16 + hi * 8;  // C/D layout: 8 consecutive elements per lane

  // ---- prologue: async prefetch of tile 0 into buffer 0 ----
  {
    unsigned l0 = (unsigned)(uintptr_t)(xb0 + lane * 8);
    uint64_t g0 = (uint64_t)(xrow + lane * 8);
    uint64_t g1 = (uint64_t)(xrow + lane * 8 + 4);
    asm volatile("global_load_async_to_lds_b128 %0, %1, off" :: "v"(l0),        "v"(g0) : "memory");
    asm volatile("global_load_async_to_lds_b128 %0, %1, off" :: "v"(l0 + 16u), "v"(g1) : "memory");
  }

  float carry = 0.0f;

  for (int p = 0; p < NPASS; ++p) {
    const int t0 = p * TILE;
    float* xb = xb0 + (p & 1) * 256;

    // ---- async prefetch next tile into the other buffer, then wait for current ----
    if (p + 1 < NPASS) {
      float* xbn = xb0 + ((p + 1) & 1) * 256;
      const int e0 = (p + 1) * TILE + lane * 8;
      const float* ga = (e0 + 4 <= T_LEN) ? (xrow + e0)     : xrow;  // clamp OOB tail
      const float* gb = (e0 + 8 <= T_LEN) ? (xrow + e0 + 4) : xrow;
      unsigned l0 = (unsigned)(uintptr_t)(xbn + lane * 8);
      uint64_t g0 = (uint64_t)ga, g1 = (uint64_t)gb;
      asm volatile("global_load_async_to_lds_b128 %0, %1, off" :: "v"(l0),        "v"(g0) : "memory");
      asm volatile("global_load_async_to_lds_b128 %0, %1, off" :: "v"(l0 + 16u), "v"(g1) : "memory");
      asm volatile("s_wait_asynccnt 0x2" ::: "memory");   // in-order: current tile is resident
    } else {
      asm volatile("s_wait_asynccnt 0x0" ::: "memory");
    }

    // ---- build B chunks: Y[i][j] = relu(x[j*16+i]) * q^(t0 + j*16 + i), K-striped ----
    const int ebase = n * 16 + c0;
    float dk = fexp2(L2Q * (float)(t0 + ebase));
    v2f bm[4];
    #pragma unroll
    for (int k = 0; k < 4; ++k) {
      const v2f xx = *(const v2f*)(xb + ebase + 4 * k);
      bm[k].x = fmaxf(xx.x, 0.0f) * dk;
      bm[k].y = fmaxf(xx.y, 0.0f) * (dk * q);
      dk *= q4;
    }

    // ---- phase 1: intra-segment inclusive prefix  P = L x Y  (L lower-tri ones) ----
    v8f cacc = {0.f,0.f,0.f,0.f,0.f,0.f,0.f,0.f};
    #pragma unroll
    for (int k = 0; k < 4; ++k) {
      v2f am;
      am.x = ((4*k + c0)     <= n) ? 1.0f : 0.0f;
      am.y = ((4*k + c0 + 1) <= n) ? 1.0f : 0.0f;
      cacc = __builtin_amdgcn_wmma_f32_16x16x4_f32(false, am, false, bm[k],
                                                   (short)0, cacc, false, false);
    }

    // ---- phase 2: cross-segment offsets  P += T_rep x SU  (SU strictly upper-tri) ----
    const float myTop = cacc[7];          // row M=15 (segment totals) lives in lanes 16..31
    #pragma unroll
    for (int k = 0; k < 4; ++k) {
      v2f am, bs;
      am.x = __shfl(myTop, 16 + 4*k + c0,     32);
      am.y = __shfl(myTop, 16 + 4*k + c0 + 1, 32);
      bs.x = ((4*k + c0)     < n) ? 1.0f : 0.0f;
      bs.y = ((4*k + c0 + 1) < n) ? 1.0f : 0.0f;
      cacc = __builtin_amdgcn_wmma_f32_16x16x4_f32(false, am, false, bs,
                                                   (short)0, cacc, false, false);
    }

    // ---- outputs: each lane owns 8 consecutive elements [eout, eout+8) of the tile ----
    if (t0 + eout + 8 <= T_LEN) {
      const v4f lo  = *(const v4f*)(xb + eout);
      const v4f hi4 = *(const v4f*)(xb + eout + 4);
      const float xs[8] = {lo.x, lo.y, lo.z, lo.w, hi4.x, hi4.y, hi4.z, hi4.w};
      float dec = fexp2(L2Q * (float)(t0 + eout));     // q^t, advanced by *q per element
      float ov[8];
      #pragma unroll
      for (int m = 0; m < 8; ++m) {
        const float xc = cacc[m] + carry;              // cumsum(relu(x)*q^t)
        const float wc = (1.0f - dec * q) * inv_s;     // closed-form cumsum(q^t)
        const float Mv = xc / (wc + EPSF);
        const float xr = fmaxf(xs[m], 0.0f);
        const float nr = xr * fexp2(-a_ * flog2(Mv + EPSF));   // x / (eps+M)^a
        ov[m] = fexp2(r_ * flog2(nr + d_)) - drc;              // (norm+d)^r - d^r
        dec *= q;
      }
      const v4f o0 = {ov[0], ov[1], ov[2], ov[3]};
      const v4f o1 = {ov[4], ov[5], ov[6], ov[7]};
      __builtin_nontemporal_store(o0, (v4f*)(orow + t0 + eout));
      __builtin_nontemporal_store(o1, (v4f*)(orow + t0 + eout + 4));
    }

    carry += __shfl(cacc[7], 31, 32);   // tile total = inclusive prefix at element 255
  }
}

extern "C" void kernel_launch(void* const* d_in, const int* in_sizes, int n_in,
                              void* d_out, int out_size, void* d_ws, size_t ws_size,
                              hipStream_t stream) {
  const float* x     = (const float*)d_in[0];
  const float* alpha = (const float*)d_in[1];
  const float* delta = (const float*)d_in[2];
  const float* rpow  = (const float*)d_in[3];
  const float* scoef = (const float*)d_in[4];
  float* out = (float*)d_out;

  const int C     = in_sizes[1];              // 128
  const int nrows = in_sizes[0] / T_LEN;      // B*C = 2048

  dim3 block(256);                            // 8 waves = 8 rows per block
  dim3 grid((nrows + 7) / 8);
  const size_t shmem = 8 * 512 * sizeof(float);  // per-wave double buffer

  pcen_wmma_scan_kernel<<<grid, block, shmem, stream>>>(
      x, alpha, delta, rpow, scoef, out, C, nrows);
}